// SigMMD_naive_15195594293658
// MI455X (gfx1250) — compile-verified
//
#include <hip/hip_runtime.h>
#include <hip/hip_bf16.h>

typedef __attribute__((ext_vector_type(2))) float v2f;
typedef __attribute__((ext_vector_type(8))) float v8f;

#define A_PATHS 64
#define M_PTS   64
#define D_DIM   8
#define P_INC   63              // increments per path
#define PAIRS   (A_PATHS * A_PATHS)     // 4096
#define NPAIR_TOTAL (3 * PAIRS)         // 12288
#define INC_LD  65              // padded LDS stride for the 64x64 inc tile

// One workgroup (256 threads = 8 waves) per (set, a, b) pair.
//  Phase 1: build dX/dY rows in LDS, WMMA f32 16x16x4 (K=8 as 2 chained) -> inc tile in LDS.
//  Phase 2: wave 0 solves the Goursat PDE with a lane-systolic wavefront (shfl_up),
//           2 passes of 32 rows, row 32 staged via LDS.
__global__ __launch_bounds__(256)
void sig_pde_pair_kernel(const float* __restrict__ X,
                         const float* __restrict__ Y,
                         float* __restrict__ results) {
    __shared__ float s_dx[M_PTS * D_DIM];     // dX rows (row 63 zeroed)
    __shared__ float s_dy[M_PTS * D_DIM];     // dY rows (row 63 zeroed)
    __shared__ float s_inc[M_PTS * INC_LD];   // inc[row][col], padded stride
    __shared__ float s_row[M_PTS];            // K[32][j+1] handoff between passes

    const int g   = blockIdx.x;
    const int set = g / PAIRS;          // 0:XX 1:YY 2:XY
    const int idx = g - set * PAIRS;
    const int a   = idx >> 6;
    const int b   = idx & 63;

    const float* P1 = (set == 1) ? Y : X;   // rows (A side)
    const float* P2 = (set == 0) ? X : Y;   // cols (B side)

    // ---- Phase 0: increments into LDS ----
    for (int i = threadIdx.x; i < M_PTS * D_DIM; i += blockDim.x) {
        const int row = i >> 3;
        const int d   = i & 7;
        float vx = 0.0f, vy = 0.0f;
        if (row < P_INC) {
            const int bx = (a * M_PTS + row) * D_DIM + d;
            const int by = (b * M_PTS + row) * D_DIM + d;
            vx = P1[bx + D_DIM] - P1[bx];
            vy = P2[by + D_DIM] - P2[by];
        }
        s_dx[i] = vx;
        s_dy[i] = vy;
    }
    __syncthreads();

    // ---- Phase 1: inc = dX (64x8) * dY^T (8x64) via WMMA f32 16x16x4 ----
    {
        const int wave = threadIdx.x >> 5;
        const int lane = threadIdx.x & 31;
        const int half = lane >> 4;       // 0: K 0..1 / M 0..7 rows of D; 1: K 2..3 / M 8..15
        const int l16  = lane & 15;

        for (int tsel = 0; tsel < 2; ++tsel) {
            const int tile = wave + tsel * 8;     // 0..15
            const int ti = tile >> 2;             // row-tile
            const int tj = tile & 3;              // col-tile
            const int m = ti * 16 + l16;          // A row for this lane
            const int n = tj * 16 + l16;          // B col for this lane

            v8f acc = {0.f, 0.f, 0.f, 0.f, 0.f, 0.f, 0.f, 0.f};
            #pragma unroll
            for (int kk = 0; kk < D_DIM; kk += 4) {
                const int kb = kk + half * 2;
                v2f Af, Bf;
                Af[0] = s_dx[m * D_DIM + kb + 0];
                Af[1] = s_dx[m * D_DIM + kb + 1];
                Bf[0] = s_dy[n * D_DIM + kb + 0];   // B[k][n] = dY[n][k]
                Bf[1] = s_dy[n * D_DIM + kb + 1];
                acc = __builtin_amdgcn_wmma_f32_16x16x4_f32(
                    /*neg_a=*/false, Af, /*neg_b=*/false, Bf,
                    /*c_mod=*/(short)0, acc, /*reuse_a=*/false, /*reuse_b=*/false);
            }
            // D layout: reg r -> M = r (lanes 0-15) / r+8 (lanes 16-31), N = lane&15
            #pragma unroll
            for (int r = 0; r < 8; ++r) {
                const int row = ti * 16 + half * 8 + r;
                const int col = tj * 16 + l16;
                s_inc[row * INC_LD + col] = acc[r];
            }
        }
    }
    __syncthreads();

    // ---- Phase 2: Goursat PDE, wave 0 only, lane-systolic wavefront ----
    if (threadIdx.x < 32) {
        const int lane = threadIdx.x;
        #pragma unroll 1
        for (int p = 0; p < 2; ++p) {
            const int r = p * 32 + lane;          // inc row owned by this lane
            const bool haveRow = (r < P_INC);
            float kleft = 1.0f;                   // K[r+1][j]   (own previous output)
            float kdiag = 1.0f;                   // K[r][j]     (lagging upper value)
            float kout  = 1.0f;                   // last produced K[r+1][j+1]
            #pragma unroll 1
            for (int t = 0; t <= 93; ++t) {       // 62 + 31 skewed steps
                float kup = __shfl_up(kout, 1);   // K[r][j+1] from lane above, step t-1
                if (lane == 0)
                    kup = (p == 0) ? 1.0f : s_row[(t < P_INC) ? t : (P_INC - 1)];
                const int j = t - lane;
                if (haveRow && j >= 0 && j < P_INC) {
                    const float incv = s_inc[r * INC_LD + j];
                    const float knew = fmaf(kdiag, incv - 1.0f, kup + kleft);
                    kdiag = kup;
                    kleft = knew;
                    kout  = knew;
                    if (p == 0 && lane == 31) s_row[j] = knew;  // K[32][j+1]
                }
            }
            // row 62 lives on lane 30 of pass 1 (p==1): kleft == K[63][63]
            if (p == 1 && lane == 30) results[g] = kleft;
        }
    }
}

// Deterministic weighted reduction:
//   out = mean(K_XX) + mean(K_YY) - 2*mean(K_XY) + mean((X[:,0,:]-Y[:,0,:])^2)
__global__ __launch_bounds__(256)
void sig_reduce_kernel(const float* __restrict__ X,
                       const float* __restrict__ Y,
                       const float* __restrict__ results,
                       float* __restrict__ out) {
    __shared__ float sbuf[256];
    float acc = 0.0f;
    for (int gIdx = threadIdx.x; gIdx < NPAIR_TOTAL; gIdx += 256) {
        const float w = (gIdx < 2 * PAIRS) ? (1.0f / (float)PAIRS)
                                           : (-2.0f / (float)PAIRS);
        acc += w * results[gIdx];
    }
    // baseline term: 512 elements (X[a,0,d] - Y[a,0,d])^2
    for (int i = threadIdx.x; i < A_PATHS * D_DIM; i += 256) {
        const int a = i >> 3;
        const int d = i & 7;
        const float diff = X[a * M_PTS * D_DIM + d] - Y[a * M_PTS * D_DIM + d];
        acc += diff * diff * (1.0f / (float)(A_PATHS * D_DIM));
    }
    sbuf[threadIdx.x] = acc;
    __syncthreads();
    for (int s = 128; s > 0; s >>= 1) {
        if (threadIdx.x < s) sbuf[threadIdx.x] += sbuf[threadIdx.x + s];
        __syncthreads();
    }
    if (threadIdx.x == 0) out[0] = sbuf[0];
}

extern "C" void kernel_launch(void* const* d_in, const int* in_sizes, int n_in,
                              void* d_out, int out_size, void* d_ws, size_t ws_size,
                              hipStream_t stream) {
    (void)in_sizes; (void)n_in; (void)out_size; (void)ws_size;
    const float* X = (const float*)d_in[0];
    const float* Y = (const float*)d_in[1];
    float* results = (float*)d_ws;          // NPAIR_TOTAL floats of scratch

    sig_pde_pair_kernel<<<dim3(NPAIR_TOTAL), dim3(256), 0, stream>>>(X, Y, results);
    sig_reduce_kernel<<<dim3(1), dim3(256), 0, stream>>>(X, Y, results, (float*)d_out);
}